// RegionProposalNetwork_66967130079762
// MI455X (gfx1250) — compile-verified
//
#include <hip/hip_runtime.h>
#include <stdint.h>

typedef __attribute__((ext_vector_type(2))) float v2f;
typedef __attribute__((ext_vector_type(8))) float v8f;
typedef __attribute__((ext_vector_type(4))) unsigned int u32x4;
typedef __attribute__((ext_vector_type(8))) int i32x8;
typedef __attribute__((ext_vector_type(4))) int i32x4;

#if defined(__has_builtin)
#if __has_builtin(__builtin_amdgcn_tensor_load_to_lds) && \
    __has_builtin(__builtin_amdgcn_s_wait_tensorcnt)
#define USE_TDM 1
#endif
#endif

#define FH 100
#define FW 100
#define NB 4
#define CIN 512
#define CMID 512
#define KTOT (512*9)          // 4608
#define NCHUNK (KTOT/64)      // 72
#define NPOS 40000            // NB*FH*FW
#define PPI  10000            // FH*FW
#define NANCH 90000           // PPI*9
#define NOUT 54               // 36 loc + 18 score
#define PRE_NMS_K 6000
#define POST_NMS_N 300
#define NEG_INF_F (-1e30f)
#define KC 64

// d_out layout (floats, concatenated in reference return order)
#define OFF_OFFSET 0          // 4*90000*4
#define OFF_SCORE  1440000    // 4*90000*2
#define OFF_ROIS   2160000    // 1200*4
#define OFF_IDX    2164800    // 1200
#define OFF_ANCH   2166000    // 90000*4   (total 2,526,000)

// ---------------- conv 3x3 (implicit GEMM, WMMA f32 16x16x4) + fused 1x1 ----
// B (weight) tiles staged to LDS via the Tensor Data Mover, double-buffered so
// the DMA overlaps the im2col A-gather and the WMMA math.
__global__ __launch_bounds__(256) void conv_wmma(
    const float* __restrict__ x, const float* __restrict__ wconv,
    const float* __restrict__ bconv,
    const float* __restrict__ wscore, const float* __restrict__ wloc,
    float* __restrict__ accum)
{
    __shared__ float As[64][KC + 1];      // 64 positions x 64 K
    __shared__ float Bs[2][32][KC + 1];   // double-buffered 32 ch x 64 K (+1 pad)
    __shared__ float Hs[64][33];          // relu'd tile for fused 1x1

    const int mt = blockIdx.x;            // 625 M-tiles of 64 positions
    const int nt = blockIdx.y;            // 16 N-tiles of 32 channels
    const int tid  = threadIdx.x;
    const int wave = tid >> 5, lane = tid & 31;
    const int wm = wave & 3, wn = wave >> 2;   // 4x2 wave grid -> 64x32 tile
    const int hf = lane >> 4, l = lane & 15;

    v8f acc = {};

#if USE_TDM
    // --- Tensor DMA descriptor (uniform / SGPR material) ---
    const unsigned ldsB0 = (unsigned)(size_t)(&Bs[0][0][0]);
    const unsigned long long gbase =
        (unsigned long long)(size_t)wconv +
        (unsigned long long)(nt * 32) * (unsigned long long)KTOT * 4ull;
    i32x8 g1;
    g1[0] = (2 << 16)      // data_size = 4 bytes
          | (1 << 20)      // pad_enable
          | (5 << 22);     // pad_interval: every 64 dwords  (pad_amount=0 -> 1 dword)
    g1[1] = (int)(4608u << 16);   // tensor_dim0[15:0] (bits 79:48 start at 48)
    g1[2] = (int)(512u  << 16);   // tensor_dim0[31:16]=0 | tensor_dim1[15:0]
    g1[3] = (int)(64u   << 16);   // tensor_dim1[31:16]=0 | tile_dim0=64
    g1[4] = 32;                   // tile_dim1=32 (tile_dim2=0)
    g1[5] = 4608;                 // tensor_dim0_stride[31:0]
    g1[6] = 0;                    // stride0 hi | stride1 lo
    g1[7] = 0;
    const i32x4 gz4 = (i32x4)0;
    const i32x8 gz8 = (i32x8)0;

    #define TDM_ISSUE(k0v, bufv)                                              \
        do {                                                                  \
            unsigned long long ga = gbase + (unsigned long long)(k0v) * 4ull; \
            u32x4 g0;                                                         \
            g0[0] = 1u;  /* count=1, user mode */                             \
            g0[1] = ldsB0 + (unsigned)(bufv) * (32u * (KC + 1) * 4u);         \
            g0[2] = (unsigned)(ga & 0xFFFFFFFFull);                           \
            g0[3] = (unsigned)(ga >> 32) | (2u << 30); /* type=2 (image) */   \
            __builtin_amdgcn_tensor_load_to_lds(g0, g1, gz4, gz4, gz8, 0);    \
        } while (0)

    if (wave == 0) TDM_ISSUE(0, 0);       // prologue: chunk 0 -> buffer 0
#endif

    for (int c = 0; c < NCHUNK; ++c) {
        const int k0 = c * KC;
        const int buf = c & 1;

        // stage A: im2col gather, 64x64 elements (overlaps in-flight TDM)
        for (int idx = tid; idx < 64 * KC; idx += 256) {
            int row = idx >> 6, kk = idx & (KC - 1);
            int k  = k0 + kk;
            int ci = k / 9; int rr = k - ci * 9;
            int ky = rr / 3; int kx = rr - ky * 3;
            int p0 = mt * 64 + row;
            int b  = p0 / PPI; int rem = p0 - b * PPI;
            int y  = rem / FW; int xx = rem - y * FW;
            int iy = y + ky - 1, ix = xx + kx - 1;
            float v = 0.f;
            if (iy >= 0 && iy < FH && ix >= 0 && ix < FW)
                v = x[((b * CIN + ci) * FH + iy) * FW + ix];
            As[row][kk] = v;
        }
#if USE_TDM
        if (wave == 0) __builtin_amdgcn_s_wait_tensorcnt(0);  // B[buf] landed
#else
        for (int idx = tid; idx < 32 * KC; idx += 256) {      // fallback copy
            int crow = idx >> 6, kk = idx & (KC - 1);
            Bs[buf][crow][kk] = wconv[(nt * 32 + crow) * KTOT + k0 + kk];
        }
#endif
        __syncthreads();
#if USE_TDM
        if (wave == 0 && c + 1 < NCHUNK)                      // prefetch next B
            TDM_ISSUE((c + 1) * KC, (c + 1) & 1);
#endif
        #pragma unroll
        for (int kb = 0; kb < KC; kb += 4) {
            // lanes 0-15: K=kb,kb+1 ; lanes 16-31: K=kb+2,kb+3 (A and B mirror)
            v2f a, bb;
            a.x  = As[wm * 16 + l][kb + 2 * hf];
            a.y  = As[wm * 16 + l][kb + 2 * hf + 1];
            bb.x = Bs[buf][wn * 16 + l][kb + 2 * hf];
            bb.y = Bs[buf][wn * 16 + l][kb + 2 * hf + 1];
            acc = __builtin_amdgcn_wmma_f32_16x16x4_f32(
                false, a, false, bb, (short)0, acc, false, false);
        }
        __syncthreads();
    }

    // bias + ReLU -> Hs   (C/D layout: vgpr r -> M=r (lanes<16) / r+8 (lanes>=16), N=l)
    {
        int co = nt * 32 + wn * 16 + l;
        float bc = bconv[co];
        #pragma unroll
        for (int r = 0; r < 8; ++r) {
            float h = acc[r] + bc;
            Hs[wm * 16 + r + 8 * hf][wn * 16 + l] = h > 0.f ? h : 0.f;
        }
    }
    __syncthreads();

    // fused 1x1 convs: partial dots over this block's 32 channels
    for (int idx = tid; idx < 64 * NOUT; idx += 256) {
        int pl = idx / NOUT, o = idx - pl * NOUT;
        const float* wv = (o < 36) ? (wloc + o * CMID + nt * 32)
                                   : (wscore + (o - 36) * CMID + nt * 32);
        float s = 0.f;
        #pragma unroll
        for (int cch = 0; cch < 32; ++cch) s += Hs[pl][cch] * wv[cch];
        atomicAdd(&accum[(mt * 64 + pl) * NOUT + o], s);
    }
}

// ---------------- bias + layout transpose + softmax fg ----------------------
__global__ void post_kernel(const float* __restrict__ accum,
                            const float* __restrict__ bloc,
                            const float* __restrict__ bscore,
                            float* __restrict__ out, float* __restrict__ fg)
{
    int p = blockIdx.x * blockDim.x + threadIdx.x;
    if (p >= NPOS) return;
    int b = p / PPI; int cell = p - b * PPI;       // cell = y*FW + x
    const float* ap = accum + p * NOUT;
    for (int a = 0; a < 9; ++a) {
        int anch = cell * 9 + a;
        int base = OFF_OFFSET + (b * NANCH + anch) * 4;
        #pragma unroll
        for (int j = 0; j < 4; ++j)
            out[base + j] = ap[a * 4 + j] + bloc[a * 4 + j];
        float s0 = ap[36 + a * 2 + 0] + bscore[a * 2 + 0];
        float s1 = ap[36 + a * 2 + 1] + bscore[a * 2 + 1];
        int sb = OFF_SCORE + (b * NANCH + anch) * 2;
        out[sb + 0] = s0; out[sb + 1] = s1;
        float m = fmaxf(s0, s1);
        float e0 = expf(s0 - m), e1 = expf(s1 - m);
        fg[b * NANCH + anch] = e1 / (e0 + e1);
    }
}

// ---------------- anchors ---------------------------------------------------
__global__ void anchors_kernel(float* __restrict__ out, const int* __restrict__ img_h)
{
    int i = blockIdx.x * blockDim.x + threadIdx.x;
    if (i >= NANCH) return;
    int a = i % 9, cell = i / 9;
    int y = cell / FW, xx = cell - y * FW;
    int ri = a / 3, si = a - ri * 3;
    const float ratios[3] = {0.5f, 1.0f, 2.0f};
    float sc = (float)(8 << si);
    float h = 16.0f * sc * sqrtf(ratios[ri]);
    float w = 16.0f * sc * sqrtf(1.0f / ratios[ri]);
    float stride = (float)img_h[0] / (float)FH;
    float sy = y * stride, sx = xx * stride;
    float* o = out + OFF_ANCH + i * 4;
    o[0] = sy + 8.f - 0.5f * h;
    o[1] = sx + 8.f - 0.5f * w;
    o[2] = sy + 8.f + 0.5f * h;
    o[3] = sx + 8.f + 0.5f * w;
}

// ---------------- loc2bbox + clip + min-size + sortable key -----------------
__global__ void propose_kernel(const float* __restrict__ out, const float* __restrict__ fg,
                               float* __restrict__ roi, float* __restrict__ sc,
                               unsigned* __restrict__ keys,
                               const int* img_h, const int* img_w, const int* scale)
{
    int b = blockIdx.y;
    int i = blockIdx.x * blockDim.x + threadIdx.x;
    if (i >= NANCH) return;
    const float* an = out + OFF_ANCH + i * 4;
    const float* lc = out + OFF_OFFSET + (b * NANCH + i) * 4;
    float a0 = an[0], a1 = an[1], a2 = an[2], a3 = an[3];
    float ah = a2 - a0, aw = a3 - a1;
    float acy = a0 + 0.5f * ah, acx = a1 + 0.5f * aw;
    float dy = lc[0], dx = lc[1], dh = lc[2], dw = lc[3];
    float cy = acy + dy * ah, cx = acx + dx * aw;
    float hh = expf(dh) * ah, ww = expf(dw) * aw;
    float H = (float)img_h[0], W = (float)img_w[0];
    float y0 = fminf(fmaxf(cy - 0.5f * hh, 0.f), H);
    float x0 = fminf(fmaxf(cx - 0.5f * ww, 0.f), W);
    float y1 = fminf(fmaxf(cy + 0.5f * hh, 0.f), H);
    float x1 = fminf(fmaxf(cx + 0.5f * ww, 0.f), W);
    int base = b * NANCH + i;
    roi[base * 4 + 0] = y0; roi[base * 4 + 1] = x0;
    roi[base * 4 + 2] = y1; roi[base * 4 + 3] = x1;
    float ms = 16.0f * (float)scale[0];
    bool ok = ((y1 - y0) >= ms) && ((x1 - x0) >= ms);
    float s = ok ? fg[base] : 2.0f * NEG_INF_F;
    sc[base] = s;
    unsigned u = __float_as_uint(s);
    keys[base] = (u & 0x80000000u) ? ~u : (u | 0x80000000u);  // descending order key
}

// ---------------- exact top-6000 threshold via 4-pass radix select ----------
__global__ void select_init(unsigned* prefix, unsigned* remain)
{
    int b = threadIdx.x;
    if (b < NB) { prefix[b] = 0u; remain[b] = PRE_NMS_K; }
}

__global__ void hist_pass(const unsigned* __restrict__ keys, const unsigned* __restrict__ prefix,
                          unsigned* __restrict__ hist, int shiftv)
{
    int b = blockIdx.y;
    unsigned pfx = prefix[b];
    for (int i = blockIdx.x * blockDim.x + threadIdx.x; i < NANCH; i += gridDim.x * blockDim.x) {
        unsigned k = keys[b * NANCH + i];
        bool cand = (shiftv == 24) || (((k ^ pfx) >> (shiftv + 8)) == 0u);
        if (cand) atomicAdd(&hist[b * 256 + ((k >> shiftv) & 0xFFu)], 1u);
    }
}

__global__ void hist_scan(unsigned* hist, unsigned* prefix, unsigned* remain, int shiftv)
{
    int b = threadIdx.x;
    if (b >= NB) return;
    unsigned rem = remain[b], pfx = prefix[b];
    for (int bin = 255; bin >= 0; --bin) {
        unsigned c = hist[b * 256 + bin];
        if (c >= rem) { pfx |= ((unsigned)bin) << shiftv; break; }
        rem -= c;
    }
    prefix[b] = pfx; remain[b] = rem;
    for (int j = 0; j < 256; ++j) hist[b * 256 + j] = 0u;
}

__global__ void init_nms(const unsigned* __restrict__ keys, const float* __restrict__ sc,
                         const unsigned* __restrict__ prefix, unsigned* __restrict__ supp)
{
    int b = blockIdx.y;
    int i = blockIdx.x * blockDim.x + threadIdx.x;
    if (i >= NANCH) return;
    int idx = b * NANCH + i;
    bool elig = (keys[idx] >= prefix[b]) && (sc[idx] > NEG_INF_F);
    supp[idx] = elig ? 0u : 1u;
}

// ---------------- greedy NMS (one workgroup per image) ----------------------
__global__ __launch_bounds__(256) void nms_kernel(const float* __restrict__ roi,
                                                  const float* __restrict__ sc,
                                                  unsigned* __restrict__ supp,
                                                  float* __restrict__ out)
{
    int b = blockIdx.x, tid = threadIdx.x;
    __shared__ float sB[256];
    __shared__ int   sI[256];
    __shared__ float box[4];
    __shared__ int   hasFlag;
    const float* R = roi + b * NANCH * 4;
    const float* S = sc  + b * NANCH;
    unsigned*    P = supp + b * NANCH;

    for (int it = 0; it < POST_NMS_N; ++it) {
        float bestS = -3.0e38f; int bestI = -1;
        for (int i = tid; i < NANCH; i += 256)
            if (P[i] == 0u) { float v = S[i]; if (v > bestS) { bestS = v; bestI = i; } }
        sB[tid] = bestS; sI[tid] = bestI;
        __syncthreads();
        for (int off = 128; off > 0; off >>= 1) {
            if (tid < off) {
                float s2 = sB[tid + off]; int i2 = sI[tid + off];
                if (i2 >= 0 && (sI[tid] < 0 || s2 > sB[tid] ||
                                (s2 == sB[tid] && i2 < sI[tid]))) {
                    sB[tid] = s2; sI[tid] = i2;
                }
            }
            __syncthreads();
        }
        if (tid == 0) {
            int idx = sI[0];
            int rb = OFF_ROIS + (b * POST_NMS_N + it) * 4;
            if (idx >= 0) {
                box[0] = R[idx * 4 + 0]; box[1] = R[idx * 4 + 1];
                box[2] = R[idx * 4 + 2]; box[3] = R[idx * 4 + 3];
                out[rb + 0] = box[0]; out[rb + 1] = box[1];
                out[rb + 2] = box[2]; out[rb + 3] = box[3];
                hasFlag = 1;
            } else {
                out[rb + 0] = 0.f; out[rb + 1] = 0.f;
                out[rb + 2] = 0.f; out[rb + 3] = 0.f;
                hasFlag = 0;
            }
            out[OFF_IDX + b * POST_NMS_N + it] = (float)b;  // roi_indices
        }
        __syncthreads();
        if (hasFlag) {
            float by0 = box[0], bx0 = box[1], by1 = box[2], bx1 = box[3];
            float areaB = (by1 - by0) * (bx1 - bx0);
            for (int i = tid; i < NANCH; i += 256) {
                if (P[i] == 0u) {
                    float y0 = R[i*4+0], x0 = R[i*4+1], y1 = R[i*4+2], x1 = R[i*4+3];
                    float yy0 = fmaxf(by0, y0), xx0 = fmaxf(bx0, x0);
                    float yy1 = fminf(by1, y1), xx1 = fminf(bx1, x1);
                    float inter = fmaxf(yy1 - yy0, 0.f) * fmaxf(xx1 - xx0, 0.f);
                    float areaI = (y1 - y0) * (x1 - x0);
                    float iou = inter / (areaI + areaB - inter + 1e-9f);
                    if (iou > 0.7f) P[i] = 1u;
                }
            }
        }
        __syncthreads();
    }
}

// ---------------- workspace zero-init ---------------------------------------
__global__ void zero_kernel(float* __restrict__ accum, unsigned* __restrict__ histArea)
{
    int i = blockIdx.x * blockDim.x + threadIdx.x;
    if (i < NPOS * NOUT) accum[i] = 0.f;
    if (i < 256 * NB + 2 * NB) histArea[i] = 0u;
}

extern "C" void kernel_launch(void* const* d_in, const int* in_sizes, int n_in,
                              void* d_out, int out_size, void* d_ws, size_t ws_size,
                              hipStream_t stream)
{
    const float* x      = (const float*)d_in[0];
    const float* wconv  = (const float*)d_in[1];
    const float* bconv  = (const float*)d_in[2];
    const float* wscore = (const float*)d_in[3];
    const float* bscore = (const float*)d_in[4];
    const float* wloc   = (const float*)d_in[5];
    const float* bloc   = (const float*)d_in[6];
    const int*   img_h  = (const int*)d_in[7];
    const int*   img_w  = (const int*)d_in[8];
    const int*   scale  = (const int*)d_in[9];
    float* out = (float*)d_out;

    // workspace layout (~20.2 MB)
    float*    accum  = (float*)d_ws;               // 2,160,000
    float*    fg     = accum + NPOS * NOUT;        //   360,000
    float*    roi    = fg + NB * NANCH;            // 1,440,000
    float*    sc     = roi + NB * NANCH * 4;       //   360,000
    unsigned* keys   = (unsigned*)(sc + NB * NANCH);  // 360,000
    unsigned* supp   = keys + NB * NANCH;          //   360,000
    unsigned* hist   = supp + NB * NANCH;          //     1,024
    unsigned* prefix = hist + 256 * NB;            //         4
    unsigned* remain = prefix + NB;                //         4

    zero_kernel<<<(NPOS * NOUT + 255) / 256, 256, 0, stream>>>(accum, hist);

    conv_wmma<<<dim3(625, 16), 256, 0, stream>>>(x, wconv, bconv, wscore, wloc, accum);

    post_kernel<<<(NPOS + 127) / 128, 128, 0, stream>>>(accum, bloc, bscore, out, fg);

    anchors_kernel<<<(NANCH + 255) / 256, 256, 0, stream>>>(out, img_h);

    propose_kernel<<<dim3((NANCH + 255) / 256, NB), 256, 0, stream>>>(
        out, fg, roi, sc, keys, img_h, img_w, scale);

    select_init<<<1, NB, 0, stream>>>(prefix, remain);
    for (int shiftv = 24; shiftv >= 0; shiftv -= 8) {
        hist_pass<<<dim3(88, NB), 256, 0, stream>>>(keys, prefix, hist, shiftv);
        hist_scan<<<1, NB, 0, stream>>>(hist, prefix, remain, shiftv);
    }

    init_nms<<<dim3((NANCH + 255) / 256, NB), 256, 0, stream>>>(keys, sc, prefix, supp);

    nms_kernel<<<NB, 256, 0, stream>>>(roi, sc, supp, out);
}